// EPPCore_81355270521006
// MI455X (gfx1250) — compile-verified
//
#include <hip/hip_runtime.h>

typedef __attribute__((ext_vector_type(4))) float v4f;  // native float4 for NT builtins

// Problem constants (fixed by the reference's setup_inputs)
#define BZ    4
#define H     192
#define W     640
#define PIX   (H * W)          // 122880 pixels per batch
#define M     200              // maxinsnum
#define CAND  16
#define CE    36               // ch*cw = 6*6
#define TBL   (M * CE)         // 7200 floats per batch table
#define TBL4  (TBL / 4)        // 1800 float4 per batch table

#define CHUNK_SCAT 1024        // pixels per block in scatter kernel
#define CHUNK_INFL 2048        // pixels per block in inflate kernel

// ---------------------------------------------------------------------------
// K1: table[b,m,:] = selsrc[b, m, batchidx[b,m], :]
// Fully overwrites the workspace table (no zero pass needed).
// ---------------------------------------------------------------------------
__global__ __launch_bounds__(256) void k_build_table(
    const int* __restrict__ batchidx,
    const float* __restrict__ selsrc,
    float* __restrict__ table)
{
  int t = blockIdx.x * 256 + threadIdx.x;     // over BZ*M*CE = 28800
  if (t >= BZ * TBL) return;
  int b = t / TBL;
  int r = t - b * TBL;
  int m = r / CE;
  int e = r - m * CE;
  int c = batchidx[b * M + m];
  table[t] = selsrc[(((size_t)(b * M + m) * CAND + c) * CE) + e];
}

// ---------------------------------------------------------------------------
// K2: segment-sum of compsrc into table, aggregated per-block in LDS.
//   Per block: 1024 pixels of one batch. 28.8 KB LDS bin table, ds_add_f32
//   atomics per element, then one global_atomic_add_f32 flush per bin.
//   compsrc is single-use -> non-temporal loads (don't pollute L2; keep L2
//   capacity for `instance`, which the inflate pass re-reads).
// ---------------------------------------------------------------------------
__global__ __launch_bounds__(256) void k_scatter(
    const int* __restrict__ instance,
    const float* __restrict__ compsrc,
    float* __restrict__ table)
{
  __shared__ float sh[TBL];                       // 28800 B
  const int blocksPerBatch = PIX / CHUNK_SCAT;    // 120
  const int b  = blockIdx.x / blocksPerBatch;
  const int c0 = (blockIdx.x % blocksPerBatch) * CHUNK_SCAT;

  for (int i = threadIdx.x; i < TBL; i += 256) sh[i] = 0.0f;
  __syncthreads();

  for (int p = c0 + threadIdx.x; p < c0 + CHUNK_SCAT; p += 256) {
    const int idx = instance[(size_t)b * PIX + p];     // RT: re-read by K3
    const v4f* src = (const v4f*)(compsrc + ((size_t)b * PIX + p) * CE);
    float* dst = sh + idx * CE;
#pragma unroll
    for (int k = 0; k < 9; ++k) {
      v4f v = __builtin_nontemporal_load(src + k);     // global_load_b128 NT
      atomicAdd(dst + 4 * k + 0, v.x);                 // ds_add_f32
      atomicAdd(dst + 4 * k + 1, v.y);
      atomicAdd(dst + 4 * k + 2, v.z);
      atomicAdd(dst + 4 * k + 3, v.w);
    }
  }
  __syncthreads();

  float* gt = table + (size_t)b * TBL;
  for (int i = threadIdx.x; i < TBL; i += 256) {
    float v = sh[i];
    if (v != 0.0f) atomicAdd(gt + i, v);               // global_atomic_add_f32
  }
}

// ---------------------------------------------------------------------------
// K3: inflate gather. Stage the 28.8 KB per-batch (comp+sel) table into LDS
//   with the gfx1250 async global->LDS path (ASYNCcnt-tracked), then gather
//   9 float4 per pixel from LDS and store b128 non-temporal to out.
// ---------------------------------------------------------------------------
__global__ __launch_bounds__(256) void k_inflate(
    const int* __restrict__ instance,
    const float* __restrict__ table,
    float* __restrict__ out)
{
  __shared__ v4f sh4[TBL4];                       // 28800 B
  const int blocksPerBatch = PIX / CHUNK_INFL;    // 60
  const int b  = blockIdx.x / blocksPerBatch;
  const int c0 = (blockIdx.x % blocksPerBatch) * CHUNK_INFL;

  const v4f* gt4 = (const v4f*)(table + (size_t)b * TBL);
  // Low 32 bits of a generic LDS pointer are the wave-relative LDS offset.
  unsigned lds_base = (unsigned)(size_t)(&sh4[0]);

  for (int i = threadIdx.x; i < TBL4; i += 256) {
    const v4f* g = gt4 + i;
    unsigned l = lds_base + 16u * (unsigned)i;
    // VDST = LDS byte address, VADDR = 64-bit global address, no saddr.
    asm volatile("global_load_async_to_lds_b128 %0, %1, off"
                 :: "v"(l), "v"(g) : "memory");
  }
  // Wait for this wave's async LDS writes, then barrier so every wave sees
  // the whole staged table.
  asm volatile("s_wait_asynccnt 0" ::: "memory");
  __syncthreads();

  for (int p = c0 + threadIdx.x; p < c0 + CHUNK_INFL; p += 256) {
    const int idx = instance[(size_t)b * PIX + p];       // should hit L2 (RT)
    const v4f* s = &sh4[idx * 9];                        // ds_load_b128 x9
    v4f* d = (v4f*)(out + ((size_t)b * PIX + p) * CE);
#pragma unroll
    for (int k = 0; k < 9; ++k) {
      v4f v = s[k];
      __builtin_nontemporal_store(v, d + k);             // global_store_b128 NT
    }
  }
}

// ---------------------------------------------------------------------------
extern "C" void kernel_launch(void* const* d_in, const int* in_sizes, int n_in,
                              void* d_out, int out_size, void* d_ws, size_t ws_size,
                              hipStream_t stream) {
  const int*   instance = (const int*)d_in[0];   // (4,1,192,640) i32
  const float* compsrc  = (const float*)d_in[1]; // (4,192,640,6,6) f32
  const int*   batchidx = (const int*)d_in[2];   // (4,200) i32
  const float* selsrc   = (const float*)d_in[3]; // (4,200,16,6,6) f32
  float*       out      = (float*)d_out;         // (4,192,640,6,6) f32
  float*       table    = (float*)d_ws;          // 4*200*36 f32 = 115200 B scratch

  k_build_table<<<(BZ * TBL + 255) / 256, 256, 0, stream>>>(batchidx, selsrc, table);
  k_scatter<<<BZ * (PIX / CHUNK_SCAT), 256, 0, stream>>>(instance, compsrc, table);
  k_inflate<<<BZ * (PIX / CHUNK_INFL), 256, 0, stream>>>(instance, table, out);
}